// _DiagonalSSMScan1D_86904368267765
// MI455X (gfx1250) — compile-verified
//
#include <hip/hip_runtime.h>
#include <cstdint>

// Problem constants (match reference)
#define B_     8
#define L_     4096
#define D_     1024
#define CHUNKS 16
#define TC     256            // timesteps per chunk (L_/CHUNKS)
#define TILE_T 16             // timesteps staged per TDM tile
#define NT     (TC / TILE_T)  // 16 tiles per chunk
#define DBLK   256            // channels per block (= blockDim.x)
#define NDBLK  (D_ / DBLK)    // 4

#if defined(__HIP_DEVICE_COMPILE__) && \
    __has_builtin(__builtin_amdgcn_tensor_load_to_lds) && \
    __has_builtin(__builtin_amdgcn_s_wait_tensorcnt)
#define HAVE_TDM 1
#endif

typedef unsigned int uint32x4 __attribute__((ext_vector_type(4)));
typedef int          int32x4  __attribute__((ext_vector_type(4)));
typedef int          int32x8  __attribute__((ext_vector_type(8)));

#ifdef HAVE_TDM
// Build a 2D TDM descriptor (D# groups per CDNA5 ISA)
// and issue TENSOR_LOAD_TO_LDS: rows x cols elements of f32, row stride
// `row_stride` elements, into LDS byte offset `lds_byte` (contiguous tile).
__device__ __forceinline__ void tdm_load_tile_2d(const float* gsrc,
                                                 unsigned lds_byte,
                                                 int rows, int cols,
                                                 int row_stride) {
  unsigned long long ga = (unsigned long long)(uintptr_t)gsrc;
  uint32x4 g0;
  g0.x = 1u;                                    // count=1 (valid user D#)
  g0.y = lds_byte;                              // lds_addr (bytes)
  g0.z = (unsigned)ga;                          // global_addr[31:0]
  g0.w = (unsigned)((ga >> 32) & 0x1FFFFFFull)  // global_addr[56:32]
         | (2u << 30);                          // type = 2 ("image")
  int32x8 g1;
  g1[0] = (int)(2u << 16);                                  // data_size=4B
  g1[1] = (int)((unsigned)(cols & 0xFFFF) << 16);           // tensor_dim0 lo16
  g1[2] = (int)(((unsigned)cols >> 16) |
                ((unsigned)(rows & 0xFFFF) << 16));         // dim0 hi | dim1 lo
  g1[3] = (int)((unsigned)(cols & 0xFFFF) << 16);           // dim1 hi(0) | tile_dim0
  g1[4] = rows;                                             // tile_dim1 | tile_dim2=0
  g1[5] = row_stride;                                       // tensor_dim0_stride lo32
  g1[6] = 0;                                                // stride hi | dim1_stride lo
  g1[7] = 0;
  int32x4 gz4 = {0, 0, 0, 0};                               // groups 2/3 unused (2D)
  int32x8 gz8 = {0, 0, 0, 0, 0, 0, 0, 0};                   // extra group (clang-23 form)
  __builtin_amdgcn_tensor_load_to_lds(g0, g1, gz4, gz4, gz8, 0);
}
#endif

// Phase 1 (WRITE_Y=false): per-(b,d,chunk) zero-init scan -> chunk partial v.
// Phase 3 (WRITE_Y=true): same scan seeded with chunk_start, emits y.
template <bool WRITE_Y>
__global__ __launch_bounds__(DBLK) void ssm_scan_chunks(
    const float* __restrict__ x, const float* __restrict__ a_logit,
    const float* __restrict__ bv, const float* __restrict__ cv,
    const float* __restrict__ dv, const float* __restrict__ chunk_start,
    float* __restrict__ chunk_partial, float* __restrict__ y) {
  const int bid   = blockIdx.x;
  const int dblk  = bid & (NDBLK - 1);
  const int chunk = (bid >> 2) & (CHUNKS - 1);
  const int b     = bid >> 6;
  const int tid   = threadIdx.x;
  const int d     = dblk * DBLK + tid;
  const int t0    = chunk * TC;

  const size_t tile_base = ((size_t)b * L_ + t0) * D_ + (size_t)dblk * DBLK;
  const float* xb = x + tile_base;

  float a = 1.0f / (1.0f + __expf(-a_logit[d]));
  a = fminf(fmaxf(a, 1e-4f), 1.0f - 1e-4f);
  const float bb = bv[d];
  float cc = 0.0f, dd = 0.0f, s;
  if (WRITE_Y) {
    cc = cv[d];
    dd = dv[d];
    s  = chunk_start[((size_t)b * CHUNKS + chunk) * D_ + d];
  } else {
    s = 0.0f;
  }

#ifdef HAVE_TDM
  __shared__ float smem[2 * TILE_T * DBLK];  // 32 KB double buffer
  const bool issuer = (tid < 32);            // wave 0 drives the TDM
  const unsigned lds_base = (unsigned)(uintptr_t)(&smem[0]);

  if (issuer)  // prime tile 0 -> buffer 0
    tdm_load_tile_2d(xb, lds_base, TILE_T, DBLK, D_);

  for (int j = 0; j < NT; ++j) {
    if (j + 1 < NT) {
      if (issuer) {
        // kick tile j+1 into the other buffer, then wait for tile j only
        tdm_load_tile_2d(xb + (size_t)(j + 1) * TILE_T * D_,
                         lds_base + (unsigned)(((j + 1) & 1) * TILE_T * DBLK * 4),
                         TILE_T, DBLK, D_);
        __builtin_amdgcn_s_wait_tensorcnt(1);  // TDM completes in-order per wave
      }
    } else {
      if (issuer) __builtin_amdgcn_s_wait_tensorcnt(0);
    }
    __syncthreads();  // publish wave-0's DMA'd tile to all waves

    const float* buf = &smem[(j & 1) * TILE_T * DBLK];
    float* yout = WRITE_Y ? (y + tile_base + (size_t)j * TILE_T * D_ + tid) : nullptr;
#pragma unroll
    for (int tt = 0; tt < TILE_T; ++tt) {
      float xv = buf[tt * DBLK + tid];     // ds_load, conflict-free banks
      s = fmaf(a, s, bb * xv);             // s = a*s + b*x
      if (WRITE_Y) yout[(size_t)tt * D_] = fmaf(cc, s, dd * xv);
    }
    __syncthreads();  // all reads of this buffer done before tile j+2 lands
  }
#else
  // Fallback: direct coalesced global loads (256 consecutive d per block-row).
  const float* xp = xb + tid;
  float* yout = WRITE_Y ? (y + tile_base + tid) : nullptr;
  for (int t = 0; t < TC; ++t) {
    float xv = xp[(size_t)t * D_];
    s = fmaf(a, s, bb * xv);
    if (WRITE_Y) yout[(size_t)t * D_] = fmaf(cc, s, dd * xv);
  }
#endif

  if (!WRITE_Y)
    chunk_partial[((size_t)b * CHUNKS + chunk) * D_ + d] = s;
}

// Phase 2: scan the 16 chunk states per (b,d): s[j+1] = a^TC * s[j] + v[j].
__global__ __launch_bounds__(256) void ssm_chunk_prefix(
    const float* __restrict__ a_logit, const float* __restrict__ chunk_partial,
    float* __restrict__ chunk_start) {
  const int gid = blockIdx.x * 256 + threadIdx.x;  // B_*D_ threads
  const int b = gid >> 10;
  const int d = gid & (D_ - 1);
  float a = 1.0f / (1.0f + __expf(-a_logit[d]));
  a = fminf(fmaxf(a, 1e-4f), 1.0f - 1e-4f);
  const float A = __expf((float)TC * __logf(a));   // a^256
  float s = 0.0f;
#pragma unroll
  for (int j = 0; j < CHUNKS; ++j) {
    const size_t idx = ((size_t)b * CHUNKS + j) * D_ + d;
    chunk_start[idx] = s;
    s = fmaf(A, s, chunk_partial[idx]);
  }
}

extern "C" void kernel_launch(void* const* d_in, const int* in_sizes, int n_in,
                              void* d_out, int out_size, void* d_ws, size_t ws_size,
                              hipStream_t stream) {
  (void)in_sizes; (void)n_in; (void)out_size; (void)ws_size;
  const float* x  = (const float*)d_in[0];
  const float* al = (const float*)d_in[1];
  const float* bv = (const float*)d_in[2];
  const float* cv = (const float*)d_in[3];
  const float* dv = (const float*)d_in[4];
  float* y = (float*)d_out;

  float* partial = (float*)d_ws;                              // B*CHUNKS*D f32
  float* start   = partial + (size_t)B_ * CHUNKS * D_;        // B*CHUNKS*D f32

  const dim3 grid(B_ * CHUNKS * NDBLK);  // 512 blocks x 256 threads
  ssm_scan_chunks<false><<<grid, DBLK, 0, stream>>>(x, al, bv, cv, dv,
                                                    nullptr, partial, nullptr);
  ssm_chunk_prefix<<<(B_ * D_) / 256, 256, 0, stream>>>(al, partial, start);
  ssm_scan_chunks<true><<<grid, DBLK, 0, stream>>>(x, al, bv, cv, dv,
                                                   start, partial, y);
}